// GCN_32272384262905
// MI455X (gfx1250) — compile-verified
//
#include <hip/hip_runtime.h>
#include <hip/hip_bf16.h>

#define NNODES 100000
#define NEDGES 3200000
#define NFEAT  512
#define NHID   256
#define NCLASS 40

typedef float v2f __attribute__((ext_vector_type(2)));
typedef float v8f __attribute__((ext_vector_type(8)));

// ---------------------------------------------------------------------------
// Zero-fill (needed before atomic-accumulate SpMM; harness poisons ws)
// ---------------------------------------------------------------------------
__global__ __launch_bounds__(256) void gcn_zero_f32(float* __restrict__ p, long n) {
    long i = (long)blockIdx.x * blockDim.x + threadIdx.x;
    long stride = (long)gridDim.x * blockDim.x;
    for (; i < n; i += stride) p[i] = 0.0f;
}

// ---------------------------------------------------------------------------
// GEMM1: support1[100000,256] = x[100000,512] @ W1[512,256]
// One wave -> 16x64 output strip (4 x v8f accumulators), K stepped by 4 with
// V_WMMA_F32_16X16X4_F32.  25000 wave-tiles = 3125 blocks x 8 waves.
// A frag (16x4 f32): lane L<16 holds A[m0+L][k0+{0,1}], L>=16 holds K+2 pair.
// B frag (4x16 f32): lane L<16 holds B[k0+{0,1}][n], L>=16 holds K+2 pair.
// D (16x16 f32): vgpr r, lane l -> m = r + 8*(l/16), n = l%16.
// ---------------------------------------------------------------------------
__global__ __launch_bounds__(256) void gcn_gemm1_wmma(const float* __restrict__ x,
                                                      const float* __restrict__ W1,
                                                      float* __restrict__ out) {
    const int lane = threadIdx.x & 31;
    const int tile = blockIdx.x * 8 + (threadIdx.x >> 5);   // 0..24999 (uniform per wave)
    const int mtile  = tile % 6250;
    const int nstrip = tile / 6250;                         // 0..3
    const int m0 = mtile * 16;
    const int n0 = nstrip * 64;
    const int lo = lane & 15;
    const int hi = lane >> 4;

    v8f acc0 = {}, acc1 = {}, acc2 = {}, acc3 = {};
    const float* arow = x + (size_t)(m0 + lo) * NFEAT + 2 * hi;

    for (int k0 = 0; k0 < NFEAT; k0 += 4) {
        v2f a;
        a.x = arow[k0];
        a.y = arow[k0 + 1];
        const int kb = k0 + 2 * hi;
        const float* bp = W1 + (size_t)kb * NHID + n0 + lo;
        v2f b0, b1, b2, b3;
        b0.x = bp[0];   b0.y = bp[NHID +  0];
        b1.x = bp[16];  b1.y = bp[NHID + 16];
        b2.x = bp[32];  b2.y = bp[NHID + 32];
        b3.x = bp[48];  b3.y = bp[NHID + 48];
        acc0 = __builtin_amdgcn_wmma_f32_16x16x4_f32(false, a, false, b0, (short)0, acc0, false, false);
        acc1 = __builtin_amdgcn_wmma_f32_16x16x4_f32(false, a, false, b1, (short)0, acc1, false, false);
        acc2 = __builtin_amdgcn_wmma_f32_16x16x4_f32(false, a, false, b2, (short)0, acc2, false, false);
        acc3 = __builtin_amdgcn_wmma_f32_16x16x4_f32(false, a, false, b3, (short)0, acc3, false, false);
    }

    float* orow = out + (size_t)(m0 + 8 * hi) * NHID + n0 + lo;
#pragma unroll
    for (int r = 0; r < 8; ++r) {
        orow[(size_t)r * NHID +  0] = acc0[r];
        orow[(size_t)r * NHID + 16] = acc1[r];
        orow[(size_t)r * NHID + 32] = acc2[r];
        orow[(size_t)r * NHID + 48] = acc3[r];
    }
}

// ---------------------------------------------------------------------------
// SpMM (F = 256): one wave per edge; coalesced float4 row read of the source
// features (102 MB operand -> resident in 192 MB L2), scale by edge weight,
// scatter with no-return global_atomic_add_f32 into out[dst].
// ---------------------------------------------------------------------------
__global__ __launch_bounds__(256) void gcn_spmm_h(const int* __restrict__ ei,
                                                  const float* __restrict__ ew,
                                                  const float* __restrict__ feat,
                                                  float* __restrict__ out) {
    const int e = blockIdx.x * 8 + (threadIdx.x >> 5);
    if (e >= NEDGES) return;
    const int lane = threadIdx.x & 31;
    const int src = ei[e];
    const int dst = ei[NEDGES + e];
    const float w = ew[e];
    const float4* fin = (const float4*)(feat + (size_t)src * NHID);
    float* o = out + (size_t)dst * NHID;
#pragma unroll
    for (int c = 0; c < 2; ++c) {
        const int f4 = c * 32 + lane;            // 64 float4 per row
        float4 v = fin[f4];
        const int f = f4 * 4;
        unsafeAtomicAdd(o + f + 0, w * v.x);
        unsafeAtomicAdd(o + f + 1, w * v.y);
        unsafeAtomicAdd(o + f + 2, w * v.z);
        unsafeAtomicAdd(o + f + 3, w * v.w);
    }
}

// ---------------------------------------------------------------------------
// h1 = relu(h1 + b1)   (in place; 25.6M elements)
// ---------------------------------------------------------------------------
__global__ __launch_bounds__(256) void gcn_bias_relu(float* __restrict__ h,
                                                     const float* __restrict__ b1) {
    const long i = (long)blockIdx.x * blockDim.x + threadIdx.x;   // exactly NNODES*NHID threads
    const float v = h[i] + b1[i & (NHID - 1)];
    h[i] = v > 0.0f ? v : 0.0f;
}

// ---------------------------------------------------------------------------
// GEMM2: support2[100000,40] = h1[100000,256] @ W2[256,40]
// One wave -> 16 rows x 48 cols (3 tiles, last half-masked to 40).
// ---------------------------------------------------------------------------
__global__ __launch_bounds__(256) void gcn_gemm2_wmma(const float* __restrict__ h,
                                                      const float* __restrict__ W2,
                                                      float* __restrict__ out) {
    const int wave = blockIdx.x * 8 + (threadIdx.x >> 5);   // uniform per wave
    if (wave >= 6250) return;                               // wave-uniform exit; EXEC stays full
    const int lane = threadIdx.x & 31;
    const int lo = lane & 15;
    const int hi = lane >> 4;
    const int m0 = wave * 16;
    const int n2 = 32 + lo;                                 // third tile column
    const int n2c = (n2 < NCLASS) ? n2 : 0;                 // clamp to stay in-bounds
    const bool v2ok = (n2 < NCLASS);

    v8f acc0 = {}, acc1 = {}, acc2 = {};
    const float* arow = h + (size_t)(m0 + lo) * NHID + 2 * hi;

    for (int k0 = 0; k0 < NHID; k0 += 4) {
        v2f a;
        a.x = arow[k0];
        a.y = arow[k0 + 1];
        const int kb = k0 + 2 * hi;
        const float* bp = W2 + (size_t)kb * NCLASS;
        v2f b0, b1, b2;
        b0.x = bp[lo];        b0.y = bp[NCLASS + lo];
        b1.x = bp[16 + lo];   b1.y = bp[NCLASS + 16 + lo];
        float t0 = bp[n2c], t1 = bp[NCLASS + n2c];
        b2.x = v2ok ? t0 : 0.0f;
        b2.y = v2ok ? t1 : 0.0f;
        acc0 = __builtin_amdgcn_wmma_f32_16x16x4_f32(false, a, false, b0, (short)0, acc0, false, false);
        acc1 = __builtin_amdgcn_wmma_f32_16x16x4_f32(false, a, false, b1, (short)0, acc1, false, false);
        acc2 = __builtin_amdgcn_wmma_f32_16x16x4_f32(false, a, false, b2, (short)0, acc2, false, false);
    }

    float* orow = out + (size_t)(m0 + 8 * hi) * NCLASS + lo;
#pragma unroll
    for (int r = 0; r < 8; ++r) {
        orow[(size_t)r * NCLASS +  0] = acc0[r];
        orow[(size_t)r * NCLASS + 16] = acc1[r];
        if (v2ok) orow[(size_t)r * NCLASS + 32] = acc2[r];
    }
}

// ---------------------------------------------------------------------------
// SpMM (F = 40): one wave per edge; lanes cover 40 columns (32 + 8).
// ---------------------------------------------------------------------------
__global__ __launch_bounds__(256) void gcn_spmm_c(const int* __restrict__ ei,
                                                  const float* __restrict__ ew,
                                                  const float* __restrict__ feat,
                                                  float* __restrict__ out) {
    const int e = blockIdx.x * 8 + (threadIdx.x >> 5);
    if (e >= NEDGES) return;
    const int lane = threadIdx.x & 31;
    const int src = ei[e];
    const int dst = ei[NEDGES + e];
    const float w = ew[e];
    const float* fin = feat + (size_t)src * NCLASS;
    float* o = out + (size_t)dst * NCLASS;
    for (int f = lane; f < NCLASS; f += 32) {
        unsafeAtomicAdd(o + f, w * fin[f]);
    }
}

// ---------------------------------------------------------------------------
// Row softmax over 40 classes, one thread per node.
// ---------------------------------------------------------------------------
__global__ __launch_bounds__(256) void gcn_softmax(const float* __restrict__ logits,
                                                   float* __restrict__ out) {
    const int node = blockIdx.x * blockDim.x + threadIdx.x;
    if (node >= NNODES) return;
    const float* l = logits + (size_t)node * NCLASS;
    float m = -3.402823466e+38f;
#pragma unroll
    for (int j = 0; j < NCLASS; ++j) m = fmaxf(m, l[j]);
    float e[NCLASS];
    float s = 0.0f;
#pragma unroll
    for (int j = 0; j < NCLASS; ++j) { e[j] = __expf(l[j] - m); s += e[j]; }
    const float inv = 1.0f / s;
    float* o = out + (size_t)node * NCLASS;
#pragma unroll
    for (int j = 0; j < NCLASS; ++j) o[j] = e[j] * inv;
}

// ---------------------------------------------------------------------------
// Orchestration. Workspace layout (bytes):
//   [0,              102,400,000) support1 (x@W1)   -- dead after spmm1
//   [102.4e6,        204,800,000) h1
//   support2 reuses offset 0         (100000*40*4 = 16 MB)
//   logits   reuses offset 16,000,000 (16 MB, still inside dead support1)
// Total ws needed: 204.8 MB.
// ---------------------------------------------------------------------------
extern "C" void kernel_launch(void* const* d_in, const int* in_sizes, int n_in,
                              void* d_out, int out_size, void* d_ws, size_t ws_size,
                              hipStream_t stream) {
    const float* x  = (const float*)d_in[0];
    const int*   ei = (const int*)d_in[1];
    const float* ew = (const float*)d_in[2];
    const float* W1 = (const float*)d_in[3];
    const float* b1 = (const float*)d_in[4];
    const float* W2 = (const float*)d_in[5];
    float* out = (float*)d_out;

    char* ws = (char*)d_ws;
    float* support1 = (float*)ws;                                       // 100000*256
    float* h1       = (float*)(ws + (size_t)NNODES * NHID * 4);         // 100000*256
    float* support2 = (float*)ws;                                       // reuse (100000*40)
    float* logits   = (float*)(ws + (size_t)NNODES * NCLASS * 4);       // reuse (100000*40)

    // 1) support1 = x @ W1           (WMMA f32)
    gcn_gemm1_wmma<<<3125, 256, 0, stream>>>(x, W1, support1);
    // 2) h1 = 0
    gcn_zero_f32<<<8192, 256, 0, stream>>>(h1, (long)NNODES * NHID);
    // 3) h1 += adj-scatter of support1
    gcn_spmm_h<<<NEDGES / 8, 256, 0, stream>>>(ei, ew, support1, h1);
    // 4) h1 = relu(h1 + b1)
    gcn_bias_relu<<<(NNODES * NHID) / 256, 256, 0, stream>>>(h1, b1);
    // 5) support2 = h1 @ W2          (WMMA f32)
    gcn_gemm2_wmma<<<782, 256, 0, stream>>>(h1, W2, support2);
    // 6) logits = 0
    gcn_zero_f32<<<4096, 256, 0, stream>>>(logits, (long)NNODES * NCLASS);
    // 7) logits += adj-scatter of support2
    gcn_spmm_c<<<NEDGES / 8, 256, 0, stream>>>(ei, ew, support2, logits);
    // 8) out = softmax(logits)
    gcn_softmax<<<(NNODES + 255) / 256, 256, 0, stream>>>(logits, out);
}